// TauLoss_14491219657065
// MI455X (gfx1250) — compile-verified
//
#include <hip/hip_runtime.h>
#include <hip/hip_bf16.h>

// Masked segment NLL reduction:
//   loss = sum_interior( log(tau) + (dt+eps)/tau ) / (B+1)
// Interior = per segment, indices [offsets[b]+1, offsets[b+1]-1).
//
// Roofline: 2 x 16.7M x 4B = 134MB read @ 23.3 TB/s => ~5.8us floor (and the
// working set fits the 192MB L2, so graph replays can run at L2 speed with
// default RT temporal hints). Inner loop trimmed to 2 trans + 3 VALU ops per
// element (v_log_f32 * ln2 via FMA, v_rcp_f32 via FMA) so issue slots are
// dominated by the coalesced global_load_b32 stream.
//
// Cross-lane reduction uses V_WMMA_F32_16X16X4_F32 (f32 A/B/C: no precision
// loss), then 2 LDS slots per wave and a single global_atomic_add_f32 per
// block (16384 atomics total).

typedef __attribute__((ext_vector_type(2))) float v2f;
typedef __attribute__((ext_vector_type(8))) float v8f;

#define BLOCK 256  // 8 wave32 waves

__global__ __launch_bounds__(BLOCK)
void TauLoss_kernel(const float* __restrict__ tau,
                    const float* __restrict__ dt,
                    const int*   __restrict__ offsets,
                    float*       __restrict__ out,
                    float inv_div) {
    const int seg = blockIdx.x;
    const int s = offsets[seg];
    const int e = offsets[seg + 1];

    const float LN2 = 0.69314718055994530942f;

    // Per-lane partial sum over the segment interior (coalesced streaming).
    float p = 0.0f;
#pragma unroll 4
    for (int i = s + 1 + (int)threadIdx.x; i < e - 1; i += BLOCK) {
        const float t = tau[i];
        const float d = dt[i];
        const float l2 = __builtin_amdgcn_logf(t);   // v_log_f32: log2(t)
        const float r  = __builtin_amdgcn_rcpf(t);   // v_rcp_f32: 1/t
        p = fmaf(l2, LN2, p);                        // + ln(t)
        p = fmaf(d + 1e-5f, r, p);                   // + (dt+eps)/t
    }

    // ---- Wave32 cross-lane reduce via V_WMMA_F32_16X16X4_F32 ----
    // A (16x4 f32): VGPR0 = lane partial (K=0 for lanes 0-15, K=2 for 16-31),
    // VGPR1 zeroed (K=1/K=3). B = all-ones 4x16. Then D[m][n] = p_m + p_{m+16}.
    // Lane L<16 holds D[0..7][L]; lane L>=16 holds D[8..15][L-16]; summing the
    // 8 D VGPRs per lane yields the two half-wave totals.
    v2f a;  a.x = p;    a.y = 0.0f;
    v2f b;  b.x = 1.0f; b.y = 1.0f;
    v8f c = {};
    c = __builtin_amdgcn_wmma_f32_16x16x4_f32(
            /*neg_a=*/false, a, /*neg_b=*/false, b,
            /*c_mod=*/(short)0, c, /*reuse_a=*/false, /*reuse_b=*/false);
    const float w = c[0] + c[1] + c[2] + c[3] + c[4] + c[5] + c[6] + c[7];

    __shared__ float sred[(BLOCK / 32) * 2];
    const int lane = threadIdx.x & 31;
    const int wid  = threadIdx.x >> 5;
    if (lane == 0)  sred[wid * 2 + 0] = w;
    if (lane == 16) sred[wid * 2 + 1] = w;
    __syncthreads();

    if (threadIdx.x == 0) {
        float tot = 0.0f;
#pragma unroll
        for (int k = 0; k < (BLOCK / 32) * 2; ++k) tot += sred[k];
        atomicAdd(out, tot * inv_div);   // global_atomic_add_f32
    }
}

extern "C" void kernel_launch(void* const* d_in, const int* in_sizes, int n_in,
                              void* d_out, int out_size, void* d_ws, size_t ws_size,
                              hipStream_t stream) {
    const float* tau     = (const float*)d_in[0];
    const float* dt      = (const float*)d_in[1];
    const int*   offsets = (const int*)d_in[2];
    float*       out     = (float*)d_out;

    const int n_off = in_sizes[2];     // B + 1
    const int n_seg = n_off - 1;       // number of segments
    const float inv_div = 1.0f / (float)n_off;

    // d_out is poisoned by the harness; zero the accumulator (capturable).
    hipMemsetAsync(d_out, 0, sizeof(float), stream);

    TauLoss_kernel<<<n_seg, BLOCK, 0, stream>>>(tau, dt, offsets, out, inv_div);
}